// EncoderLayer_32478542693017
// MI455X (gfx1250) — compile-verified
//
#include <hip/hip_runtime.h>

typedef __attribute__((ext_vector_type(2))) float v2f;
typedef __attribute__((ext_vector_type(8))) float v8f;

#define N_NODES 50000
#define N_EDGES 800000
#define HIDDEN  64
#define LN_EPS  1e-5f
#define GEN_EPS 1e-7f
#define SM_EPS  1e-16f

// ---------------------------------------------------------------- zero fill
__global__ __launch_bounds__(256) void zero_kernel(float* __restrict__ p, int n) {
    int i = blockIdx.x * 256 + threadIdx.x;
    if (i < n) p[i] = 0.0f;
}

// ------------------------------------------------------- LayerNorm, 1 wave/node
__global__ __launch_bounds__(256) void ln_kernel(const float* __restrict__ x,
                                                 const float* __restrict__ gamma,
                                                 const float* __restrict__ beta,
                                                 float* __restrict__ y) {
    int wave = (blockIdx.x * 256 + threadIdx.x) >> 5;     // node id
    int lane = threadIdx.x & 31;
    if (wave >= N_NODES) return;

    const float2* xr = (const float2*)(x + (size_t)wave * HIDDEN);
    float2 v = xr[lane];

    float s = v.x + v.y;
    #pragma unroll
    for (int off = 16; off; off >>= 1) s += __shfl_xor(s, off, 32);
    float mu = s * (1.0f / HIDDEN);

    float dx = v.x - mu, dy = v.y - mu;
    float q = dx * dx + dy * dy;
    #pragma unroll
    for (int off = 16; off; off >>= 1) q += __shfl_xor(q, off, 32);
    float rstd = rsqrtf(q * (1.0f / HIDDEN) + LN_EPS);

    float2 g = ((const float2*)gamma)[lane];
    float2 b = ((const float2*)beta)[lane];
    float2 o;
    o.x = dx * rstd * g.x + b.x;
    o.y = dy * rstd * g.y + b.y;
    ((float2*)(y + (size_t)wave * HIDDEN))[lane] = o;
}

// -------------------------------------------- edge pass 1: segment max (1 wave/edge)
// msg > 0 always (relu + 1e-7), so uint-max on the bit pattern == float max,
// and a 0.0f-initialized buffer reproduces the reference's empty-segment -> 0.
__global__ __launch_bounds__(256) void edge_max_kernel(
        const float* __restrict__ y, const long long* __restrict__ ei,
        const float* __restrict__ w, const float* __restrict__ w_edge,
        const float* __restrict__ b_edge, unsigned int* __restrict__ segmax) {
    int e    = (blockIdx.x * 256 + threadIdx.x) >> 5;
    int lane = threadIdx.x & 31;
    if (e >= N_EDGES) return;

    long long s = ei[e];
    long long d = ei[N_EDGES + e];
    float wv = w[e];

    float2 yv = ((const float2*)(y + (size_t)s * HIDDEN))[lane];
    float2 we = ((const float2*)w_edge)[lane];
    float2 be = ((const float2*)b_edge)[lane];

    float m0 = fmaxf(yv.x + wv * we.x + be.x, 0.0f) + GEN_EPS;
    float m1 = fmaxf(yv.y + wv * we.y + be.y, 0.0f) + GEN_EPS;

    unsigned int* p = segmax + (size_t)d * HIDDEN + 2 * lane;
    atomicMax(p,     __float_as_uint(m0));
    atomicMax(p + 1, __float_as_uint(m1));
}

// --------------------- edge pass 2: accumulate denom = sum(ex), num = sum(ex*msg)
__global__ __launch_bounds__(256) void edge_sum_kernel(
        const float* __restrict__ y, const long long* __restrict__ ei,
        const float* __restrict__ w, const float* __restrict__ w_edge,
        const float* __restrict__ b_edge, const float* __restrict__ segmax,
        float* __restrict__ denom, float* __restrict__ num) {
    int e    = (blockIdx.x * 256 + threadIdx.x) >> 5;
    int lane = threadIdx.x & 31;
    if (e >= N_EDGES) return;

    long long s = ei[e];
    long long d = ei[N_EDGES + e];
    float wv = w[e];

    float2 yv = ((const float2*)(y + (size_t)s * HIDDEN))[lane];
    float2 we = ((const float2*)w_edge)[lane];
    float2 be = ((const float2*)b_edge)[lane];

    float m0 = fmaxf(yv.x + wv * we.x + be.x, 0.0f) + GEN_EPS;
    float m1 = fmaxf(yv.y + wv * we.y + be.y, 0.0f) + GEN_EPS;

    size_t base = (size_t)d * HIDDEN + 2 * lane;
    float2 mx = *(const float2*)(segmax + base);
    float ex0 = expf(m0 - mx.x);
    float ex1 = expf(m1 - mx.y);

    atomicAdd(denom + base,     ex0);
    atomicAdd(denom + base + 1, ex1);
    atomicAdd(num   + base,     ex0 * m0);
    atomicAdd(num   + base + 1, ex1 * m1);
}

// ------------------------ finalize: z = y + num/(denom+eps), in-place into y
__global__ __launch_bounds__(256) void finalize_kernel(float* __restrict__ y,
                                                       const float* __restrict__ num,
                                                       const float* __restrict__ denom) {
    int i = blockIdx.x * 256 + threadIdx.x;
    if (i < N_NODES * HIDDEN) y[i] = y[i] + num[i] / (denom[i] + SM_EPS);
}

// ---------------- out = relu(z @ w_mlp.T + b_mlp), fp32 WMMA 16x16x4
// One wave per 16x16 output tile; K=64 -> 16 WMMA steps.
// A (16x4): lane<16 -> M=lane, K={k0,k0+1}; lane>=16 -> M=lane-16, K={k0+2,k0+3}
// B (4x16): lane<16 -> N=lane, K={k0,k0+1}; lane>=16 -> N=lane-16, K={k0+2,k0+3}
// C/D:      VGPR v, lane<16 -> (M=v,   N=lane); lane>=16 -> (M=v+8, N=lane-16)
__global__ __launch_bounds__(128) void mlp_wmma_kernel(
        const float* __restrict__ z, const float* __restrict__ w_mlp,
        const float* __restrict__ b_mlp, float* __restrict__ out) {
    int wave = (blockIdx.x * 128 + threadIdx.x) >> 5;
    int lane = threadIdx.x & 31;
    int mt = wave >> 2;          // 3125 M tiles (50000 = 3125*16 exactly)
    int nt = wave & 3;           // 4 N tiles
    if (mt >= N_NODES / 16) return;

    int half = lane >> 4;        // 0: K pair {0,1}; 1: K pair {2,3}
    int l16  = lane & 15;
    int arow = mt * 16 + l16;    // node row for A
    int bcol = nt * 16 + l16;    // output column for B

    const float* za = z     + (size_t)arow * HIDDEN + 2 * half;
    const float* wb = w_mlp + (size_t)bcol * HIDDEN + 2 * half;  // B[k][n] = w_mlp[n][k]

    v8f c = {};
    #pragma unroll
    for (int k0 = 0; k0 < HIDDEN; k0 += 4) {
        v2f a, b;
        a.x = za[k0]; a.y = za[k0 + 1];
        b.x = wb[k0]; b.y = wb[k0 + 1];
        c = __builtin_amdgcn_wmma_f32_16x16x4_f32(false, a, false, b,
                                                  (short)0, c, false, false);
    }

    float bias = b_mlp[bcol];
    #pragma unroll
    for (int v = 0; v < 8; ++v) {
        int row = mt * 16 + v + half * 8;
        float r = c[v] + bias;
        out[(size_t)row * HIDDEN + bcol] = fmaxf(r, 0.0f);
    }
}

// --------------------------------------------------------------------------
extern "C" void kernel_launch(void* const* d_in, const int* in_sizes, int n_in,
                              void* d_out, int out_size, void* d_ws, size_t ws_size,
                              hipStream_t stream) {
    const float*     x      = (const float*)d_in[0];
    const long long* ei     = (const long long*)d_in[1];   // int64 [2, E]
    const float*     w      = (const float*)d_in[2];
    const float*     ln_g   = (const float*)d_in[3];
    const float*     ln_b   = (const float*)d_in[4];
    const float*     w_edge = (const float*)d_in[5];
    const float*     b_edge = (const float*)d_in[6];
    const float*     w_mlp  = (const float*)d_in[7];
    const float*     b_mlp  = (const float*)d_in[8];
    float*           out    = (float*)d_out;

    const int NH = N_NODES * HIDDEN;
    float* y      = (float*)d_ws;        // 12.8 MB, becomes z in-place
    float* segmax = y + NH;              // 12.8 MB
    float* denom  = segmax + NH;         // 12.8 MB
    float* num    = denom + NH;          // 12.8 MB

    // 1. zero segmax/denom/num (contiguous)
    zero_kernel<<<(3 * NH + 255) / 256, 256, 0, stream>>>(segmax, 3 * NH);
    // 2. LayerNorm: 1 wave per node, 8 waves per block
    ln_kernel<<<(N_NODES + 7) / 8, 256, 0, stream>>>(x, ln_g, ln_b, y);
    // 3. segment max over edges
    edge_max_kernel<<<(N_EDGES + 7) / 8, 256, 0, stream>>>(
        y, ei, w, w_edge, b_edge, (unsigned int*)segmax);
    // 4. segment sums (denominator and ex*msg numerator)
    edge_sum_kernel<<<(N_EDGES + 7) / 8, 256, 0, stream>>>(
        y, ei, w, w_edge, b_edge, segmax, denom, num);
    // 5. z = y + agg (in-place)
    finalize_kernel<<<(NH + 255) / 256, 256, 0, stream>>>(y, num, denom);
    // 6. WMMA fp32 GEMM + bias + relu: 3125*4 tiles, 4 waves/block
    mlp_wmma_kernel<<<(N_NODES / 16 * 4 + 3) / 4, 128, 0, stream>>>(
        y, w_mlp, b_mlp, out);
}